// OptNet_26147760898260
// MI455X (gfx1250) — compile-verified
//
#include <hip/hip_runtime.h>
#include <hip/hip_bf16.h>

// ---------------------------------------------------------------------------
// OptNet QP (qpth pdipm, neq=0) on MI455X / gfx1250.
// Strategy: precompute Q^{-1} once (Cholesky + triangular inverse + WMMA GEMM),
// so every cho_solve is a [128,512]x[512,512] WMMA GEMM; solve the per-batch
// KKT Schur systems S_b = S0 + diag(1/d_b) with batched CG whose matvec is a
// single WMMA GEMM (S0 shared) + elementwise diagonal term. f32 WMMA
// (v_wmma_f32_16x16x4_f32) everywhere: IPM needs f32, all working data is
// L2-resident (~12MB << 192MB), so matrix throughput is the only roofline.
// NN GEMMs stage the shared B operand through LDS with the Tensor Data Mover
// (tensor_load_to_lds + s_wait_tensorcnt, double buffered) and read WMMA
// B-fragments via ds_load, removing the redundant per-wave VMEM B traffic.
// ---------------------------------------------------------------------------

typedef float v2f __attribute__((ext_vector_type(2)));
typedef float v8f __attribute__((ext_vector_type(8)));
typedef unsigned int u32x4 __attribute__((ext_vector_type(4)));
typedef int i32x4 __attribute__((ext_vector_type(4)));
typedef int i32x8 __attribute__((ext_vector_type(8)));

static constexpr int NB    = 128;   // batch
static constexpr int NZ    = 512;   // nz == nineq
static constexpr int ITERS = 20;    // qpth maxIter
static constexpr int NCG   = 16;    // fixed CG iterations per KKT solve
static constexpr float EPS_Q = 1e-4f;
static constexpr float BIG   = 1e10f;

__device__ inline unsigned int rfl(unsigned int v) {
    return (unsigned int)__builtin_amdgcn_readfirstlane((int)v);
}

// ------------------------- WMMA GEMM (direct, NT/TN) ------------------------
// C[M,N] = alpha * op(A)[M,K] * op(B)[K,N]; wave tile 16x64, 4 waves -> 256 cols.
template <int TA, int TB, bool ACC>
__global__ void __launch_bounds__(128)
k_gemm(const float* __restrict__ A, const float* __restrict__ B,
       float* __restrict__ C, int K, int lda, int ldb, int ldc, float alpha)
{
    const int wave = threadIdx.x >> 5;
    const int lane = threadIdx.x & 31;
    const int half = lane >> 4;        // selects K-pair / M+8 per ISA layout
    const int l16  = lane & 15;
    const int mBase = blockIdx.y * 16;
    const int nBase = blockIdx.x * 256 + wave * 64;

    v8f acc[4] = {};
    const int arow = mBase + l16;

#pragma unroll 2
    for (int k0 = 0; k0 < K; k0 += 4) {
        const int kb = k0 + half * 2;
        v2f a;
        if (TA) { a.x = A[kb * lda + arow]; a.y = A[(kb + 1) * lda + arow]; }
        else    { a.x = A[arow * lda + kb]; a.y = A[arow * lda + kb + 1];   }
#pragma unroll
        for (int t = 0; t < 4; ++t) {
            const int col = nBase + t * 16 + l16;
            v2f b;
            if (TB) { b.x = B[col * ldb + kb]; b.y = B[col * ldb + kb + 1]; }
            else    { b.x = B[kb * ldb + col]; b.y = B[(kb + 1) * ldb + col]; }
            acc[t] = __builtin_amdgcn_wmma_f32_16x16x4_f32(
                false, a, false, b, (short)0, acc[t], false, false);
        }
    }
#pragma unroll
    for (int t = 0; t < 4; ++t) {
        const int col = nBase + t * 16 + l16;
#pragma unroll
        for (int r = 0; r < 8; ++r) {
            const int row = mBase + half * 8 + r;
            const float v = alpha * acc[t][r];
            if (ACC) C[row * ldc + col] += v;
            else     C[row * ldc + col]  = v;
        }
    }
}

// --------------------- WMMA GEMM, NN, TDM-staged B panels -------------------
// C[M,512] (+)= alpha * A[M,K] * B[K,512], B row stride fixed at 512 (all NN
// call sites). Block computes a 16x256 tile; B panels of KT x 256 f32 (16KB)
// are DMA'd into LDS by wave 0 via tensor_load_to_lds, double buffered; all
// waves consume them through ds_load for the WMMA B fragments.
template <bool ACC>
__global__ void __launch_bounds__(128)
k_gemm_nn_tdm(const float* __restrict__ A, const float* __restrict__ B,
              float* __restrict__ C, int K, int lda, int ldc, float alpha)
{
    constexpr int KT = 16;                      // K panel height
    __shared__ float sB[2][KT * 256];           // 2 x 16KB
    const int wave = threadIdx.x >> 5;
    const int lane = threadIdx.x & 31;
    const int half = lane >> 4;
    const int l16  = lane & 15;
    const int mBase = blockIdx.y * 16;
    const int nBlk  = blockIdx.x * 256;
    const int nOff  = wave * 64;

    const unsigned int ldsBase =
        rfl((unsigned int)(uintptr_t)(void*)&sB[0][0]);

    auto issue = [&](int kp, int bufSel) {
        const unsigned long long gaddr =
            (unsigned long long)(uintptr_t)(B + (size_t)kp * KT * 512 + nBlk);
        // ---- D# group 0: count=1 | lds_addr | global_addr[56:0] | type=2
        u32x4 g0;
        g0.x = 1u;
        g0.y = ldsBase + (unsigned int)bufSel * (KT * 256 * 4);
        g0.z = rfl((unsigned int)(gaddr & 0xffffffffu));
        g0.w = rfl((unsigned int)((gaddr >> 32) & 0x01ffffffu)) | 0x80000000u;
        // ---- D# group 1: data_size=4B, dims/strides/tile
        i32x8 g1;
        g1[0] = 0x00020000;                 // workgroup_mask=0, data_size=2(4B)
        g1[1] = (512 & 0xffff) << 16;       // tensor_dim0[15:0] @ bits 63:48
        g1[2] = (512 & 0xffff) << 16;       // dim0 hi=0 | tensor_dim1[15:0]
        g1[3] = 256 << 16;                  // dim1 hi=0 | tile_dim0=256
        g1[4] = KT;                         // tile_dim1=KT, tile_dim2=0
        g1[5] = 512;                        // tensor_dim0_stride[31:0]
        g1[6] = 0;
        g1[7] = 0;
        const i32x4 gz4 = {0, 0, 0, 0};     // 2-D tensor: groups 2/3 unused
        const i32x8 gz8 = {0, 0, 0, 0, 0, 0, 0, 0};
        __builtin_amdgcn_tensor_load_to_lds(g0, g1, gz4, gz4, gz8, 0);
    };

    const int NP = K / KT;
    if (threadIdx.x < 32) issue(0, 0);

    v8f acc[4] = {};
    const int arow = mBase + l16;
    for (int kp = 0; kp < NP; ++kp) {
        if (threadIdx.x < 32) __builtin_amdgcn_s_wait_tensorcnt(0);
        __syncthreads();                       // panel kp visible to all waves
        if (threadIdx.x < 32 && kp + 1 < NP) issue(kp + 1, (kp + 1) & 1);
        const float* bp = &sB[kp & 1][0];
#pragma unroll 2
        for (int k4 = 0; k4 < KT; k4 += 4) {
            const int kg = kp * KT + k4 + half * 2;
            v2f a; a.x = A[arow * lda + kg]; a.y = A[arow * lda + kg + 1];
            const int kl = k4 + half * 2;
#pragma unroll
            for (int t = 0; t < 4; ++t) {
                const int c = nOff + t * 16 + l16;
                v2f b; b.x = bp[kl * 256 + c]; b.y = bp[(kl + 1) * 256 + c];
                acc[t] = __builtin_amdgcn_wmma_f32_16x16x4_f32(
                    false, a, false, b, (short)0, acc[t], false, false);
            }
        }
        __syncthreads();                       // done reading buf before reuse
    }
#pragma unroll
    for (int t = 0; t < 4; ++t) {
        const int col = nBlk + nOff + t * 16 + l16;
#pragma unroll
        for (int r = 0; r < 8; ++r) {
            const int row = mBase + half * 8 + r;
            const float v = alpha * acc[t][r];
            if (ACC) C[row * ldc + col] += v;
            else     C[row * ldc + col]  = v;
        }
    }
}

// ------------------------- once-per-launch scalar kernels -------------------
__global__ void k_ewmul(float* __restrict__ o, const float* __restrict__ a,
                        const float* __restrict__ b, int n) {
    int i = blockIdx.x * blockDim.x + threadIdx.x;
    if (i < n) o[i] = a[i] * b[i];
}
__global__ void k_add_eps_diag(float* __restrict__ Q, int n, float eps) {
    int j = blockIdx.x * blockDim.x + threadIdx.x;
    if (j < n) Q[j * n + j] += eps;
}
__global__ void k_cholesky(float* __restrict__ A, int n) {
    const int t = threadIdx.x;
    for (int j = 0; j < n; ++j) {
        if (t == j) A[j * n + j] = sqrtf(A[j * n + j]);
        __syncthreads();
        const float djj = A[j * n + j];
        if (t > j) A[t * n + j] /= djj;
        __syncthreads();
        if (t > j) {
            const float lij = A[t * n + j];
            for (int k = j + 1; k <= t; ++k) A[t * n + k] -= lij * A[k * n + j];
        }
        __syncthreads();
    }
    for (int k = t + 1; k < n; ++k) A[t * n + k] = 0.0f;
}
__global__ void k_trinv(const float* __restrict__ Lc, float* __restrict__ Li, int n) {
    int j = blockIdx.x * blockDim.x + threadIdx.x;
    if (j >= n) return;
    for (int i = 0; i < j; ++i) Li[i * n + j] = 0.0f;
    Li[j * n + j] = 1.0f / Lc[j * n + j];
    for (int i = j + 1; i < n; ++i) {
        float s = 0.0f;
        for (int k = j; k < i; ++k) s += Lc[i * n + k] * Li[k * n + j];
        Li[i * n + j] = -s / Lc[i * n + i];
    }
}
__global__ void k_compute_h(const float* __restrict__ G, const float* __restrict__ z0,
                            const float* __restrict__ s0, float* __restrict__ h, int n) {
    int j = blockIdx.x * blockDim.x + threadIdx.x;
    if (j >= n) return;
    float acc = s0[j];
    for (int k = 0; k < n; ++k) acc += G[j * n + k] * z0[k];
    h[j] = acc;
}
__global__ void k_init_state(float* __restrict__ bsum) { bsum[0] = 3.4e38f; }

// ------------------------------ elementwise ---------------------------------
__global__ void k_copy(float* __restrict__ d, const float* __restrict__ s, int n) {
    int i = blockIdx.x * blockDim.x + threadIdx.x; if (i < n) d[i] = s[i];
}
__global__ void k_negcopy(float* __restrict__ d, const float* __restrict__ s, int n) {
    int i = blockIdx.x * blockDim.x + threadIdx.x; if (i < n) d[i] = -s[i];
}
__global__ void k_fill(float* __restrict__ d, float v, int n) {
    int i = blockIdx.x * blockDim.x + threadIdx.x; if (i < n) d[i] = v;
}
__global__ void k_add_inplace(float* __restrict__ a, const float* __restrict__ b, int n) {
    int i = blockIdx.x * blockDim.x + threadIdx.x; if (i < n) a[i] += b[i];
}
__global__ void k_div(float* __restrict__ o, const float* __restrict__ a,
                      const float* __restrict__ b, int n) {
    int i = blockIdx.x * blockDim.x + threadIdx.x; if (i < n) o[i] = a[i] / b[i];
}
__global__ void k_rz_finish(float* __restrict__ rz, const float* __restrict__ s,
                            const float* __restrict__ h, int n) {
    int i = blockIdx.x * blockDim.x + threadIdx.x;
    if (i < n) rz[i] += s[i] - h[i & (NZ - 1)];
}
__global__ void k_hv_init(float* __restrict__ hv, const float* __restrict__ h, int n) {
    int i = blockIdx.x * blockDim.x + threadIdx.x;
    if (i < n) hv[i] += h[i & (NZ - 1)];          // rs=0, rz=-h, d=1
}
__global__ void k_hv_finish(float* __restrict__ hv, const float* __restrict__ z,
                            const float* __restrict__ d, const float* __restrict__ rz, int n) {
    int i = blockIdx.x * blockDim.x + threadIdx.x;
    if (i < n) hv[i] += z[i] / d[i] - rz[i];      // rs = z
}
__global__ void k_hv2(float* __restrict__ hv, const float* __restrict__ rsc,
                      const float* __restrict__ d, int n) {
    int i = blockIdx.x * blockDim.x + threadIdx.x;
    if (i < n) hv[i] = rsc[i] / d[i];
}
__global__ void k_ds_aff(float* __restrict__ ds, const float* __restrict__ z,
                         const float* __restrict__ dz, const float* __restrict__ d, int n) {
    int i = blockIdx.x * blockDim.x + threadIdx.x;
    if (i < n) ds[i] = (-z[i] - dz[i]) / d[i];
}
__global__ void k_ds_corr(float* __restrict__ ds, const float* __restrict__ rsc,
                          const float* __restrict__ dzc, const float* __restrict__ d, int n) {
    int i = blockIdx.x * blockDim.x + threadIdx.x;
    if (i < n) ds[i] += (-rsc[i] - dzc[i]) / d[i];
}
__global__ void k_update(float* __restrict__ v, const float* __restrict__ dv,
                         const float* __restrict__ alpha, int n) {
    int i = blockIdx.x * blockDim.x + threadIdx.x;
    if (i < n) v[i] += alpha[i >> 9] * dv[i];     // NZ == 512
}
__global__ void k_best_copy(const float* __restrict__ flag, float* __restrict__ out,
                            const float* __restrict__ x, int n) {
    if (((const int*)flag)[0] == 0) return;
    int i = blockIdx.x * blockDim.x + threadIdx.x;
    if (i < n) out[i] = x[i];
}

// ------------------------- wave-per-row reductions --------------------------
__device__ inline float wsum(float v) {
#pragma unroll
    for (int o = 16; o; o >>= 1) v += __shfl_xor(v, o, 32);
    return v;
}
__device__ inline float wmin(float v) {
#pragma unroll
    for (int o = 16; o; o >>= 1) v = fminf(v, __shfl_xor(v, o, 32));
    return v;
}
__global__ void __launch_bounds__(128)
k_resid_row(const float* __restrict__ rx, const float* __restrict__ rz,
            const float* __restrict__ s, const float* __restrict__ z,
            float* __restrict__ mu, float* __restrict__ resids) {
    const int gw = (blockIdx.x * 128 + threadIdx.x) >> 5;
    const int lane = threadIdx.x & 31;
    if (gw >= NB) return;
    const float *px = rx + gw * NZ, *pz = rz + gw * NZ, *ps = s + gw * NZ, *pq = z + gw * NZ;
    float a = 0, b = 0, c = 0;
    for (int j = lane; j < NZ; j += 32) {
        float vx = px[j], vz = pz[j];
        a += vx * vx; b += vz * vz; c += ps[j] * pq[j];
    }
    a = wsum(a); b = wsum(b); c = wsum(c);
    if (lane == 0) { mu[gw] = fabsf(c) / NZ; resids[gw] = sqrtf(b) + sqrtf(a) + fabsf(c); }
}
__global__ void k_reduce_best(const float* __restrict__ resids, float* __restrict__ bsum,
                              float* __restrict__ impF) {
    const int tid = threadIdx.x, lane = tid & 31, wv = tid >> 5;
    float v = (tid < NB) ? resids[tid] : 0.0f;
    v = wsum(v);
    __shared__ float part[4];
    if (lane == 0) part[wv] = v;
    __syncthreads();
    if (tid == 0) {
        const float rsum = part[0] + part[1] + part[2] + part[3];
        const float bs = bsum[0];
        int imp = (rsum < bs) || (bs >= 1e37f);   // force first iteration
        if (rsum < bs) bsum[0] = rsum;
        ((int*)impF)[0] = imp;
    }
}
__global__ void __launch_bounds__(128) k_shift(float* __restrict__ v) {
    const int gw = (blockIdx.x * 128 + threadIdx.x) >> 5;
    const int lane = threadIdx.x & 31;
    if (gw >= NB) return;
    float* p = v + gw * NZ;
    float mn = BIG;
    for (int j = lane; j < NZ; j += 32) mn = fminf(mn, p[j]);
    mn = wmin(mn);
    if (mn < 0.0f) for (int j = lane; j < NZ; j += 32) p[j] = p[j] - mn + 1.0f;
}
__global__ void __launch_bounds__(128)
k_alpha(const float* __restrict__ z, const float* __restrict__ dz,
        const float* __restrict__ s, const float* __restrict__ ds,
        float* __restrict__ alpha, float scale) {
    const int gw = (blockIdx.x * 128 + threadIdx.x) >> 5;
    const int lane = threadIdx.x & 31;
    if (gw >= NB) return;
    const float *pz = z + gw * NZ, *pdz = dz + gw * NZ, *ps = s + gw * NZ, *pds = ds + gw * NZ;
    float mn = BIG;
    for (int j = lane; j < NZ; j += 32) {
        float d1 = pdz[j]; if (d1 < 0.0f) mn = fminf(mn, -pz[j] / d1);
        float d2 = pds[j]; if (d2 < 0.0f) mn = fminf(mn, -ps[j] / d2);
    }
    mn = wmin(mn);
    if (lane == 0) alpha[gw] = fminf(scale * mn, 1.0f);
}
__global__ void __launch_bounds__(128)
k_sig_rsc(const float* __restrict__ s, const float* __restrict__ z,
          const float* __restrict__ ds, const float* __restrict__ dz,
          const float* __restrict__ alpha, const float* __restrict__ mu,
          float* __restrict__ rsc) {
    const int gw = (blockIdx.x * 128 + threadIdx.x) >> 5;
    const int lane = threadIdx.x & 31;
    if (gw >= NB) return;
    const float *ps = s + gw * NZ, *pz = z + gw * NZ, *pa = ds + gw * NZ, *pb = dz + gw * NZ;
    const float a = alpha[gw], m = mu[gw];
    float num = 0, den = 0;
    for (int j = lane; j < NZ; j += 32) {
        num += (ps[j] + a * pa[j]) * (pz[j] + a * pb[j]);
        den += ps[j] * pz[j];
    }
    num = wsum(num); den = wsum(den);
    const float t = num / den, sig = t * t * t;
    float* pr = rsc + gw * NZ;
    for (int j = lane; j < NZ; j += 32) pr[j] = (-m * sig + pa[j] * pb[j]) / ps[j];
}
// ------------------------------- batched CG ---------------------------------
__global__ void __launch_bounds__(128)
k_cg_init(float* __restrict__ x, float* __restrict__ r, float* __restrict__ p,
          float* __restrict__ rr, const float* __restrict__ b) {
    const int gw = (blockIdx.x * 128 + threadIdx.x) >> 5;
    const int lane = threadIdx.x & 31;
    if (gw >= NB) return;
    float acc = 0;
    for (int j = lane; j < NZ; j += 32) {
        const int i = gw * NZ + j;
        const float bv = b[i];
        x[i] = 0.0f; r[i] = bv; p[i] = bv; acc += bv * bv;
    }
    acc = wsum(acc);
    if (lane == 0) rr[gw] = acc;
}
__global__ void __launch_bounds__(128)
k_cg_step1(float* __restrict__ Ap, const float* __restrict__ p,
           const float* __restrict__ d, float* __restrict__ pAp) {
    const int gw = (blockIdx.x * 128 + threadIdx.x) >> 5;
    const int lane = threadIdx.x & 31;
    if (gw >= NB) return;
    float acc = 0;
    for (int j = lane; j < NZ; j += 32) {
        const int i = gw * NZ + j;
        const float ap = Ap[i] + p[i] / d[i];
        Ap[i] = ap; acc += p[i] * ap;
    }
    acc = wsum(acc);
    if (lane == 0) pAp[gw] = acc;
}
__global__ void __launch_bounds__(128)
k_cg_step2(float* __restrict__ x, float* __restrict__ r, float* __restrict__ p,
           const float* __restrict__ Ap, float* __restrict__ rr,
           const float* __restrict__ pAp) {
    const int gw = (blockIdx.x * 128 + threadIdx.x) >> 5;
    const int lane = threadIdx.x & 31;
    if (gw >= NB) return;
    const float rrv = rr[gw];
    const float al = rrv / fmaxf(pAp[gw], 1e-30f);
    float acc = 0;
    for (int j = lane; j < NZ; j += 32) {
        const int i = gw * NZ + j;
        x[i] += al * p[i];
        const float rv = r[i] - al * Ap[i];
        r[i] = rv; acc += rv * rv;
    }
    acc = wsum(acc);
    const float be = acc / fmaxf(rrv, 1e-30f);
    for (int j = lane; j < NZ; j += 32) {
        const int i = gw * NZ + j;
        p[i] = r[i] + be * p[i];
    }
    if (lane == 0) rr[gw] = acc;
}

// ------------------------------- host side ----------------------------------
static inline void gemm(hipStream_t st, const float* A, const float* B, float* C,
                        int M, int N, int K, int lda, int ldb, int ldc,
                        float alpha, int ta, int tb, int acc) {
    dim3 g(N / 256, M / 16), blk(128);
    if (!ta && !tb) {                 // NN: TDM-staged (all NN sites have ldb==512)
        if (!acc) k_gemm_nn_tdm<false><<<g, blk, 0, st>>>(A, B, C, K, lda, ldc, alpha);
        else      k_gemm_nn_tdm<true ><<<g, blk, 0, st>>>(A, B, C, K, lda, ldc, alpha);
    } else if (tb) {
        k_gemm<0, 1, false><<<g, blk, 0, st>>>(A, B, C, K, lda, ldb, ldc, alpha);
    } else {
        k_gemm<1, 0, false><<<g, blk, 0, st>>>(A, B, C, K, lda, ldb, ldc, alpha);
    }
}

extern "C" void kernel_launch(void* const* d_in, const int* in_sizes, int n_in,
                              void* d_out, int out_size, void* d_ws, size_t ws_size,
                              hipStream_t stream) {
    (void)in_sizes; (void)n_in; (void)out_size; (void)ws_size;
    const float* xIn = (const float*)d_in[0];   // [128,512]
    const float* Mm  = (const float*)d_in[1];   // [512,512]
    const float* Ll  = (const float*)d_in[2];   // [512,512]
    const float* G   = (const float*)d_in[3];   // [512,512]
    const float* s0  = (const float*)d_in[4];   // [512]
    const float* z0  = (const float*)d_in[5];   // [512]
    float* out = (float*)d_out;                 // zhat [128,512]

    const int MAT = NZ * NZ, BN = NB * NZ;
    float* w = (float*)d_ws;
    auto take = [&](int n) { float* p = w; w += n; return p; };
    float *Lm = take(MAT), *Q = take(MAT), *Lc = take(MAT), *Li = take(MAT);
    float *Qi = take(MAT), *GQi = take(MAT), *S0 = take(MAT);
    float *h = take(NZ);
    float *x_ = take(BN), *s_ = take(BN), *z_ = take(BN);
    float *rx = take(BN), *rz = take(BN), *dd = take(BN);
    float *iqr = take(BN), *hv = take(BN), *dz = take(BN), *ds = take(BN);
    float *dx = take(BN), *tmpB = take(BN), *rsc = take(BN), *dzc = take(BN);
    float *cgx = take(BN), *cgr = take(BN), *cgp = take(BN), *cgap = take(BN);
    float *mu = take(NB), *resids = take(NB), *alphaB = take(NB);
    float *rrB = take(NB), *pApB = take(NB);
    float *bsum = take(1), *impF = take(1);

    auto ew = [&](int n) { return dim3((n + 255) / 256); };
    const dim3 EWB(256), RB(128), RG(NB / 4);   // wave-per-row: 32 blocks x 4 waves

    auto cg_solve = [&](const float* b) {       // solve (S0 + diag(1/dd)) u = b -> cgx
        k_cg_init<<<RG, RB, 0, stream>>>(cgx, cgr, cgp, rrB, b);
        for (int i = 0; i < NCG; ++i) {
            gemm(stream, cgp, S0, cgap, NB, NZ, NZ, NZ, NZ, NZ, 1.0f, 0, 0, 0);
            k_cg_step1<<<RG, RB, 0, stream>>>(cgap, cgp, dd, pApB);
            k_cg_step2<<<RG, RB, 0, stream>>>(cgx, cgr, cgp, cgap, rrB, pApB);
        }
    };

    // ---- setup: Q, chol, Q^{-1}, S0, h -------------------------------------
    k_ewmul<<<ew(MAT), EWB, 0, stream>>>(Lm, Mm, Ll, MAT);
    gemm(stream, Lm, Lm, Q, NZ, NZ, NZ, NZ, NZ, NZ, 1.0f, 0, 1, 0);   // Q = Lm Lm^T
    k_add_eps_diag<<<ew(NZ), EWB, 0, stream>>>(Q, NZ, EPS_Q);
    k_copy<<<ew(MAT), EWB, 0, stream>>>(Lc, Q, MAT);
    k_cholesky<<<1, NZ, 0, stream>>>(Lc, NZ);
    k_trinv<<<4, 128, 0, stream>>>(Lc, Li, NZ);
    gemm(stream, Li, Li, Qi, NZ, NZ, NZ, NZ, NZ, NZ, 1.0f, 1, 0, 0);  // Qi = Li^T Li
    gemm(stream, G, Qi, GQi, NZ, NZ, NZ, NZ, NZ, NZ, 1.0f, 0, 0, 0);  // G Q^{-1}
    gemm(stream, GQi, G, S0, NZ, NZ, NZ, NZ, NZ, NZ, 1.0f, 0, 1, 0);  // S0 = GQi G^T
    k_compute_h<<<2, 256, 0, stream>>>(G, z0, s0, h, NZ);
    k_init_state<<<1, 1, 0, stream>>>(bsum);

    // ---- initial point (d=1, rx=p, rs=0, rz=-h) ----------------------------
    gemm(stream, xIn, Qi, iqr, NB, NZ, NZ, NZ, NZ, NZ, 1.0f, 0, 0, 0);
    gemm(stream, iqr, G, hv, NB, NZ, NZ, NZ, NZ, NZ, 1.0f, 0, 1, 0);
    k_hv_init<<<ew(BN), EWB, 0, stream>>>(hv, h, BN);
    k_fill<<<ew(BN), EWB, 0, stream>>>(dd, 1.0f, BN);
    cg_solve(hv);
    k_negcopy<<<ew(BN), EWB, 0, stream>>>(z_, cgx, BN);               // z = -u
    k_copy<<<ew(BN), EWB, 0, stream>>>(s_, cgx, BN);                  // s = -z = u
    k_copy<<<ew(BN), EWB, 0, stream>>>(tmpB, xIn, BN);
    gemm(stream, z_, G, tmpB, NB, NZ, NZ, NZ, NZ, NZ, 1.0f, 0, 0, 1); // tmp = p + dz@G
    gemm(stream, tmpB, Qi, x_, NB, NZ, NZ, NZ, NZ, NZ, -1.0f, 0, 0, 0);
    k_shift<<<RG, RB, 0, stream>>>(s_);
    k_shift<<<RG, RB, 0, stream>>>(z_);

    // ---- main IPM loop ------------------------------------------------------
    for (int it = 0; it < ITERS; ++it) {
        // residuals
        gemm(stream, x_, Q, rx, NB, NZ, NZ, NZ, NZ, NZ, 1.0f, 0, 0, 0);
        gemm(stream, z_, G, rx, NB, NZ, NZ, NZ, NZ, NZ, 1.0f, 0, 0, 1);
        k_add_inplace<<<ew(BN), EWB, 0, stream>>>(rx, xIn, BN);       // + p
        gemm(stream, x_, G, rz, NB, NZ, NZ, NZ, NZ, NZ, 1.0f, 0, 1, 0);
        k_rz_finish<<<ew(BN), EWB, 0, stream>>>(rz, s_, h, BN);
        k_resid_row<<<RG, RB, 0, stream>>>(rx, rz, s_, z_, mu, resids);
        k_reduce_best<<<1, 128, 0, stream>>>(resids, bsum, impF);
        k_best_copy<<<ew(BN), EWB, 0, stream>>>(impF, out, x_, BN);
        k_div<<<ew(BN), EWB, 0, stream>>>(dd, z_, s_, BN);

        // affine (predictor)
        gemm(stream, rx, Qi, iqr, NB, NZ, NZ, NZ, NZ, NZ, 1.0f, 0, 0, 0);
        gemm(stream, iqr, G, hv, NB, NZ, NZ, NZ, NZ, NZ, 1.0f, 0, 1, 0);
        k_hv_finish<<<ew(BN), EWB, 0, stream>>>(hv, z_, dd, rz, BN);
        cg_solve(hv);
        k_negcopy<<<ew(BN), EWB, 0, stream>>>(dz, cgx, BN);
        k_ds_aff<<<ew(BN), EWB, 0, stream>>>(ds, z_, dz, dd, BN);
        k_copy<<<ew(BN), EWB, 0, stream>>>(tmpB, rx, BN);
        gemm(stream, dz, G, tmpB, NB, NZ, NZ, NZ, NZ, NZ, 1.0f, 0, 0, 1);
        gemm(stream, tmpB, Qi, dx, NB, NZ, NZ, NZ, NZ, NZ, -1.0f, 0, 0, 0);
        k_alpha<<<RG, RB, 0, stream>>>(z_, dz, s_, ds, alphaB, 1.0f);
        k_sig_rsc<<<RG, RB, 0, stream>>>(s_, z_, ds, dz, alphaB, mu, rsc);

        // corrector (rx=0, rz=0, rs=rsc)
        k_hv2<<<ew(BN), EWB, 0, stream>>>(hv, rsc, dd, BN);
        cg_solve(hv);
        k_negcopy<<<ew(BN), EWB, 0, stream>>>(dzc, cgx, BN);
        k_ds_corr<<<ew(BN), EWB, 0, stream>>>(ds, rsc, dzc, dd, BN);
        gemm(stream, dzc, G, tmpB, NB, NZ, NZ, NZ, NZ, NZ, 1.0f, 0, 0, 0);
        gemm(stream, tmpB, Qi, dx, NB, NZ, NZ, NZ, NZ, NZ, -1.0f, 0, 0, 1);
        k_add_inplace<<<ew(BN), EWB, 0, stream>>>(dz, dzc, BN);

        // step
        k_alpha<<<RG, RB, 0, stream>>>(z_, dz, s_, ds, alphaB, 0.999f);
        k_update<<<ew(BN), EWB, 0, stream>>>(x_, dx, alphaB, BN);
        k_update<<<ew(BN), EWB, 0, stream>>>(s_, ds, alphaB, BN);
        k_update<<<ew(BN), EWB, 0, stream>>>(z_, dz, alphaB, BN);
    }
}